// LatencyCoding_1297080123579
// MI455X (gfx1250) — compile-verified
//
#include <hip/hip_runtime.h>

// Latency coding expand: in [32,3,224,224] f32 -> out [32,32,3,224,224] f32.
// Memory-bound: 19 MB read + 616 MB written. Strategy: one float4 of input per
// thread, build 32-bit one-hot latency masks, emit 32 fully-unrolled
// non-temporal global_store_b128 (output is 3.2x the 192MB L2 -> bypass).

typedef float v4f __attribute__((ext_vector_type(4)));

#define TIMESTEPS 32
#define SPATIAL   150528            // 3*224*224
#define S4        (SPATIAL / 4)     // 37632 float4s per (b, t) plane
#define BATCH     32
#define N4        (BATCH * S4)      // 1204224 float4s of input

__global__ __launch_bounds__(256)
void latency_code_kernel(const float* __restrict__ in, float* __restrict__ out) {
    int tid = blockIdx.x * blockDim.x + threadIdx.x;
    if (tid >= N4) return;

    const v4f* in4 = (const v4f*)in;
    v4f x = __builtin_nontemporal_load(&in4[tid]);

    // Per-component one-hot latency mask: bit t set iff spike at timestep t.
    unsigned m[4];
#pragma unroll
    for (int j = 0; j < 4; ++j) {
        float xn = x[j];
        xn = fminf(fmaxf(xn, 0.0f), 1.0f);            // clip(x, 0, 1)
        int lat = (int)((1.0f - xn) * 31.0f);         // trunc toward zero, >= 0
        lat = min(max(lat, 0), TIMESTEPS - 1);
        m[j] = (xn > 0.0f) ? (1u << lat) : 0u;
    }

    int b  = tid / S4;
    int s4 = tid - b * S4;
    v4f* out4 = (v4f*)out;
    size_t base = (size_t)b * (size_t)(TIMESTEPS * S4) + (size_t)s4;

#pragma unroll
    for (int t = 0; t < TIMESTEPS; ++t) {
        v4f sp;
        sp.x = (float)((m[0] >> t) & 1u);
        sp.y = (float)((m[1] >> t) & 1u);
        sp.z = (float)((m[2] >> t) & 1u);
        sp.w = (float)((m[3] >> t) & 1u);
        // Streaming output: non-temporal so 616MB doesn't thrash 192MB L2.
        __builtin_nontemporal_store(sp, &out4[base + (size_t)t * S4]);
    }
}

extern "C" void kernel_launch(void* const* d_in, const int* in_sizes, int n_in,
                              void* d_out, int out_size, void* d_ws, size_t ws_size,
                              hipStream_t stream) {
    (void)in_sizes; (void)n_in; (void)out_size; (void)d_ws; (void)ws_size;
    const float* x = (const float*)d_in[0];
    float* out = (float*)d_out;

    dim3 block(256);                       // 8 wave32s
    dim3 grid((N4 + 255) / 256);           // 4704 blocks, exact cover
    hipLaunchKernelGGL(latency_code_kernel, grid, block, 0, stream, x, out);
}